// SimpleAttentiveFPGNN_40235253629199
// MI455X (gfx1250) — compile-verified
//
#include <hip/hip_runtime.h>
#include <cmath>

#define NN     40000
#define NE     640000
#define HIDD   128
#define NHEAD  4
#define DHEAD  32
#define NLAYER 3

typedef __attribute__((ext_vector_type(2))) float v2f;
typedef __attribute__((ext_vector_type(4))) float v4f;
typedef __attribute__((ext_vector_type(8))) float v8f;
typedef __attribute__((ext_vector_type(4))) int   i4v;

#if defined(__has_builtin)
#if __has_builtin(__builtin_amdgcn_global_load_async_to_lds_b128) && \
    __has_builtin(__builtin_amdgcn_s_wait_asynccnt)
#define USE_ASYNC_LDS 1
#else
#define USE_ASYNC_LDS 0
#endif
#else
#define USE_ASYNC_LDS 0
#endif

__device__ __forceinline__ float sigmoidf_(float x) { return 1.0f / (1.0f + expf(-x)); }

// float atomic max via sign-aware integer atomics (init target to -inf)
__device__ __forceinline__ void atomicMaxF(float* addr, float v) {
  if (v >= 0.0f) atomicMax((int*)addr, __float_as_int(v));
  else           atomicMin((unsigned int*)addr, __float_as_uint(v));
}

// Stage a 16x128 f32 tile into LDS (two b128 per thread). Uses the gfx1250
// async-to-LDS path when the builtin exists; falls back to plain LDS stores.
__device__ __forceinline__ void stage_tile_16x128(const float* __restrict__ X,
                                                  float* lds, int nodeBase, int tid) {
  const v4f* s4 = reinterpret_cast<const v4f*>(X + (size_t)nodeBase * HIDD);
  v4f* d4 = reinterpret_cast<v4f*>(lds);
#if USE_ASYNC_LDS
  // builtin expects (int4 addrspace(1)*, int4 addrspace(3)*, i32 offset, i32 cpol)
  typedef __attribute__((address_space(1))) i4v* gp_t;
  typedef __attribute__((address_space(3))) i4v* lp_t;
  gp_t g0 = (gp_t)(unsigned long long)(s4 + tid);
  gp_t g1 = (gp_t)(unsigned long long)(s4 + tid + 256);
  lp_t l0 = (lp_t)(unsigned int)(unsigned long long)(d4 + tid);
  lp_t l1 = (lp_t)(unsigned int)(unsigned long long)(d4 + tid + 256);
  __builtin_amdgcn_global_load_async_to_lds_b128(g0, l0, 0, 0);
  __builtin_amdgcn_global_load_async_to_lds_b128(g1, l1, 0, 0);
  __builtin_amdgcn_s_wait_asynccnt(0);   // per-wave; barrier follows at call site
#else
  d4[tid]       = s4[tid];
  d4[tid + 256] = s4[tid + 256];
#endif
}

// ---------------------------------------------------------------------------
// GEMM  Y(16n x 128) = X(16n x 128) @ W(128 x 128) [+ bias], fp32 WMMA 16x16x4
// grid = NN/16 blocks, 256 threads (8 waves); wave w owns cols [16w,16w+16)
// ---------------------------------------------------------------------------
__global__ __launch_bounds__(256)
void gemm_nn_wmma(const float* __restrict__ X, const float* __restrict__ W,
                  const float* __restrict__ bias, float* __restrict__ Y) {
  __shared__ float lds[16 * 128];
  const int tid = threadIdx.x;
  const int nodeBase = blockIdx.x * 16;
  __builtin_prefetch(W + tid * 64, 0, 0);          // warm W (global_prefetch_b8)
  stage_tile_16x128(X, lds, nodeBase, tid);
  __syncthreads();

  const int wave = tid >> 5;
  const int lane = tid & 31;
  const int m    = lane & 15;     // A row / B col / D col
  const int hi   = lane >> 4;     // K sub-slot (A/B), row offset pair (D)
  const int colBase = wave * 16;

  v8f acc = {};
#pragma unroll 8
  for (int k = 0; k < 128; k += 4) {
    const int ka = k + 2 * hi;
    v2f a, b;
    a.x = lds[m * 128 + ka];
    a.y = lds[m * 128 + ka + 1];
    b.x = W[(size_t)ka * HIDD + colBase + m];
    b.y = W[(size_t)(ka + 1) * HIDD + colBase + m];
    acc = __builtin_amdgcn_wmma_f32_16x16x4_f32(false, a, false, b,
                                                (short)0, acc, false, false);
  }
  const float bv = bias ? bias[colBase + m] : 0.0f;
#pragma unroll
  for (int r = 0; r < 8; ++r) {
    const int row = r + 8 * hi;   // D layout: vgpr r -> M=r (lanes<16), M=r+8 (lanes>=16)
    Y[(size_t)(nodeBase + row) * HIDD + colBase + m] = acc[r] + bv;
  }
}

// ---------------------------------------------------------------------------
// el/er: one wave per (node, head); 32-lane dot + shfl reduction
// ---------------------------------------------------------------------------
__global__ __launch_bounds__(256)
void eler_kernel(const float* __restrict__ feat, const float* __restrict__ al,
                 const float* __restrict__ ar, float* __restrict__ el,
                 float* __restrict__ er) {
  const int wid  = blockIdx.x * 8 + (threadIdx.x >> 5);   // (node,head)
  const int lane = threadIdx.x & 31;
  const int node = wid >> 2;
  const int head = wid & 3;
  const float f  = feat[(size_t)node * HIDD + head * DHEAD + lane];
  float vl = f * al[head * DHEAD + lane];
  float vr = f * ar[head * DHEAD + lane];
#pragma unroll
  for (int o = 16; o > 0; o >>= 1) {
    vl += __shfl_xor(vl, o, 32);
    vr += __shfl_xor(vr, o, 32);
  }
  if (lane == 0) { el[wid] = vl; er[wid] = vr; }
}

__global__ __launch_bounds__(256)
void init_layer(float* __restrict__ rst, float* __restrict__ emax,
                float* __restrict__ denom) {
  const int i = blockIdx.x * 256 + threadIdx.x;
  rst[i] = 0.0f;                          // grid covers NN*HIDD exactly
  if (i < NN * NHEAD) { emax[i] = -__builtin_inff(); denom[i] = 0.0f; }
}

// pass 1: e = leaky_relu(el[src]+er[dst]); segment max via atomics
__global__ __launch_bounds__(256)
void edge_attn_max(const int* __restrict__ src, const int* __restrict__ dst,
                   const float* __restrict__ el, const float* __restrict__ er,
                   float* __restrict__ e_out, float* __restrict__ emax) {
  const int e = blockIdx.x * 256 + threadIdx.x;
  if (e >= NE) return;
  const int s = src[e], d = dst[e];
  const v4f l4 = ((const v4f*)el)[s];
  const v4f r4 = ((const v4f*)er)[d];
  v4f ev;
#pragma unroll
  for (int h = 0; h < NHEAD; ++h) {
    const float x = l4[h] + r4[h];
    const float v = x > 0.0f ? x : 0.2f * x;
    ev[h] = v;
    atomicMaxF(&emax[d * NHEAD + h], v);
  }
  ((v4f*)e_out)[e] = ev;
}

// pass 2: ee = exp(e - emax[dst]); segment sum
__global__ __launch_bounds__(256)
void edge_softmax_sum(const int* __restrict__ dst, float* __restrict__ e_io,
                      const float* __restrict__ emax, float* __restrict__ denom) {
  const int e = blockIdx.x * 256 + threadIdx.x;
  if (e >= NE) return;
  const int d = dst[e];
  v4f ev = ((const v4f*)e_io)[e];
  const v4f mx = ((const v4f*)emax)[d];
#pragma unroll
  for (int h = 0; h < NHEAD; ++h) {
    const float o = expf(ev[h] - mx[h]);
    ev[h] = o;
    atomicAdd(&denom[d * NHEAD + h], o);
  }
  ((v4f*)e_io)[e] = ev;
}

// pass 3: rst[dst] += feat[src] * alpha   (one wave per edge; lane owns 4 feats)
__global__ __launch_bounds__(256)
void edge_scatter(const int* __restrict__ src, const int* __restrict__ dst,
                  const float* __restrict__ e_s, const float* __restrict__ denom,
                  const float* __restrict__ feat, float* __restrict__ rst) {
  const int e    = blockIdx.x * 8 + (threadIdx.x >> 5);
  const int lane = threadIdx.x & 31;
  const int s = src[e], d = dst[e];
  const int head = lane >> 3;                        // 4 feats per lane -> head = lane/8
  const float alpha = e_s[e * NHEAD + head] / denom[d * NHEAD + head];
  const v4f f = ((const v4f*)feat)[(size_t)s * 32 + lane];
  float* base = rst + (size_t)d * HIDD + lane * 4;
  atomicAdd(base + 0, f[0] * alpha);
  atomicAdd(base + 1, f[1] * alpha);
  atomicAdd(base + 2, f[2] * alpha);
  atomicAdd(base + 3, f[3] * alpha);
}

__global__ __launch_bounds__(256)
void finalize_layer(const float* __restrict__ rst, const float* __restrict__ h_in,
                    const float* __restrict__ bias, float* __restrict__ h_out) {
  const int i = blockIdx.x * 256 + threadIdx.x;     // grid covers NN*HIDD exactly
  const float x = rst[i] + h_in[i] + bias[i & (HIDD - 1)];
  h_out[i] = x > 0.0f ? x : expf(x) - 1.0f;         // ELU(alpha=1)
}

// ---------------------------------------------------------------------------
// Fused GRU step: gi = X @ Wih^T + bih (WMMA into LDS), then gate math.
// h0 == 0 so Whh vanishes; only bhh survives (matches reference).
// ---------------------------------------------------------------------------
__global__ __launch_bounds__(256)
void gru_fused_wmma(const float* __restrict__ X, const float* __restrict__ Wih,
                    const float* __restrict__ bih, const float* __restrict__ bhh,
                    float* __restrict__ Y) {
  __shared__ float ldsx[16 * 128];
  __shared__ float ldsg[16 * 384];
  const int tid = threadIdx.x;
  const int nodeBase = blockIdx.x * 16;
  __builtin_prefetch(Wih + tid * 192, 0, 0);
  stage_tile_16x128(X, ldsx, nodeBase, tid);
  __syncthreads();

  const int wave = tid >> 5;
  const int lane = tid & 31;
  const int m    = lane & 15;
  const int hi   = lane >> 4;

  // 24 col-tiles of 16 over 384 gate outputs; 3 per wave
#pragma unroll
  for (int t = 0; t < 3; ++t) {
    const int colBase = (wave * 3 + t) * 16;
    v8f acc = {};
#pragma unroll 8
    for (int k = 0; k < 128; k += 4) {
      const int ka = k + 2 * hi;
      v2f a, b;
      a.x = ldsx[m * 128 + ka];
      a.y = ldsx[m * 128 + ka + 1];
      // B[k][j] = Wih[j*128 + k]  (Wih is (384,128) row-major, used transposed)
      b.x = Wih[(size_t)(colBase + m) * 128 + ka];
      b.y = Wih[(size_t)(colBase + m) * 128 + ka + 1];
      acc = __builtin_amdgcn_wmma_f32_16x16x4_f32(false, a, false, b,
                                                  (short)0, acc, false, false);
    }
    const float bv = bih[colBase + m];
#pragma unroll
    for (int r = 0; r < 8; ++r) {
      const int row = r + 8 * hi;
      ldsg[row * 384 + colBase + m] = acc[r] + bv;
    }
  }
  __syncthreads();

  // gate math: 16*128 outputs, 8 per thread
  const int row = tid >> 4;
  const int c0  = (tid & 15) * 8;
#pragma unroll
  for (int j = 0; j < 8; ++j) {
    const int c = c0 + j;
    const float r  = sigmoidf_(ldsg[row * 384 + c]       + bhh[c]);
    const float z  = sigmoidf_(ldsg[row * 384 + 128 + c] + bhh[128 + c]);
    const float nn = tanhf(ldsg[row * 384 + 256 + c] + r * bhh[256 + c]);
    Y[(size_t)(nodeBase + row) * HIDD + c] = (1.0f - z) * nn;
  }
}

// ---------------------------------------------------------------------------
extern "C" void kernel_launch(void* const* d_in, const int* in_sizes, int n_in,
                              void* d_out, int out_size, void* d_ws, size_t ws_size,
                              hipStream_t stream) {
  (void)in_sizes; (void)n_in; (void)out_size; (void)ws_size;

  const float* node_feats = (const float*)d_in[0];
  const int*   src        = (const int*)  d_in[1];
  const int*   dst        = (const int*)  d_in[2];
  const float* proj_W     = (const float*)d_in[3];
  const float* proj_b     = (const float*)d_in[4];
  const float* fc_W       = (const float*)d_in[5];
  const float* attn_l     = (const float*)d_in[6];
  const float* attn_r     = (const float*)d_in[7];
  const float* conv_bias  = (const float*)d_in[8];
  const float* gru_Wih    = (const float*)d_in[9];
  const float* gru_bih    = (const float*)d_in[11];
  const float* gru_bhh    = (const float*)d_in[12];

  char* ws = (char*)d_ws;
  size_t off = 0;
  auto alloc = [&](size_t bytes) -> void* {
    void* p = ws + off;
    off = (off + bytes + 255) & ~(size_t)255;
    return p;
  };
  float* hA    = (float*)alloc(sizeof(float) * NN * HIDD);   // 20.5 MB
  float* feat  = (float*)alloc(sizeof(float) * NN * HIDD);   // 20.5 MB (reused as hB)
  float* rst   = (float*)alloc(sizeof(float) * NN * HIDD);   // 20.5 MB
  float* el    = (float*)alloc(sizeof(float) * NN * NHEAD);
  float* er    = (float*)alloc(sizeof(float) * NN * NHEAD);
  float* emax  = (float*)alloc(sizeof(float) * NN * NHEAD);
  float* denom = (float*)alloc(sizeof(float) * NN * NHEAD);
  float* esc   = (float*)alloc(sizeof(float) * NE * NHEAD);  // 10.2 MB

  const int GB = NN / 16;            // 2500 GEMM blocks (exact tiling)
  const int NB = (NN * HIDD) / 256;  // 20000 elementwise blocks (exact)

  // projection
  gemm_nn_wmma<<<GB, 256, 0, stream>>>(node_feats, proj_W, proj_b, hA);

  for (int l = 0; l < NLAYER; ++l) {
    const float* Wl = fc_W + (size_t)l * HIDD * HIDD;
    gemm_nn_wmma<<<GB, 256, 0, stream>>>(hA, Wl, nullptr, feat);
    eler_kernel<<<(NN * NHEAD) / 8, 256, 0, stream>>>(
        feat, attn_l + l * NHEAD * DHEAD, attn_r + l * NHEAD * DHEAD, el, er);
    init_layer<<<NB, 256, 0, stream>>>(rst, emax, denom);
    edge_attn_max<<<(NE + 255) / 256, 256, 0, stream>>>(src, dst, el, er, esc, emax);
    edge_softmax_sum<<<(NE + 255) / 256, 256, 0, stream>>>(dst, esc, emax, denom);
    edge_scatter<<<NE / 8, 256, 0, stream>>>(src, dst, esc, denom, feat, rst);
    finalize_layer<<<NB, 256, 0, stream>>>(rst, hA, conv_bias + l * HIDD, hA);
  }

  // GRU x2 (h0 = 0 each step); feat buffer is dead -> reuse as ping-pong
  float* hB = feat;
  gru_fused_wmma<<<GB, 256, 0, stream>>>(hA, gru_Wih, gru_bih, gru_bhh, hB);
  gru_fused_wmma<<<GB, 256, 0, stream>>>(hB, gru_Wih, gru_bih, gru_bhh, (float*)d_out);
}